// Clustering_58557584114228
// MI455X (gfx1250) — compile-verified
//
#include <hip/hip_runtime.h>
#include <math.h>

typedef __attribute__((ext_vector_type(2))) float v2f;
typedef __attribute__((ext_vector_type(8))) float v8f;
typedef int v4i __attribute__((vector_size(16)));   // matches builtin prototype

#define GLOBAL_AS __attribute__((address_space(1)))
#define LDS_AS    __attribute__((address_space(3)))

#define BB 8
#define NN 32768
#define KK 128
#define DD 64
#define TAU 1.0f
#define ITERS 10
#define EPSV 1e-8f

#define KT 8                 // 128/16 cluster tiles
#define PTS_PER_BLOCK 128    // 8 waves x 16 points
#define NCHUNK 8
#define CHUNK (NN / NCHUNK)  // 4096
#define SUBN 128             // feats staging tile (n-extent)
#define SUBNP (SUBN + 4)     // padded LDS row stride (banks)

#if __has_builtin(__builtin_amdgcn_global_load_async_to_lds_b128)
#define HAVE_ASYNC_LDS 1
#endif

// workspace layout (float offsets)
#define WS_DENOM   0                          // BB*KK
#define WS_CNUM    (WS_DENOM + BB*KK)         // BB*KK*3
#define WS_CENT    (WS_CNUM + BB*KK*3)        // BB*KK*4 (x,y,z,pad)
#define WS_SSUM    (WS_CENT + BB*KK*4)        // BB (padded to 64)
#define WS_FACC    (WS_SSUM + 64)             // BB*KK*DD

__global__ void zero_f32(float* __restrict__ p, int n) {
    int t = blockIdx.x * blockDim.x + threadIdx.x;
    if (t < n) p[t] = 0.0f;
}

__global__ void score_sum(const float* __restrict__ s, float* __restrict__ ssum) {
    __shared__ float red[256];
    int b = blockIdx.x;
    float acc = 0.0f;
    for (int n = threadIdx.x; n < NN; n += 256) acc += s[b * NN + n];
    red[threadIdx.x] = acc;
    __syncthreads();
    for (int off = 128; off; off >>= 1) {
        if (threadIdx.x < off) red[threadIdx.x] += red[threadIdx.x + off];
        __syncthreads();
    }
    if (threadIdx.x == 0) ssum[b] = red[0] + EPSV;
}

__global__ void init_centers(const float* __restrict__ xyz, float* __restrict__ cent) {
    int t = blockIdx.x * blockDim.x + threadIdx.x;
    if (t >= BB * KK) return;
    int b = t / KK, k = t % KK;
    int n = k * (NN / KK);
    const float* xb = xyz + (size_t)b * 3 * NN;
    cent[t * 4 + 0] = xb[0 * NN + n];
    cent[t * 4 + 1] = xb[1 * NN + n];
    cent[t * 4 + 2] = xb[2 * NN + n];
    cent[t * 4 + 3] = 0.0f;
}

// One block: 8 waves x 16 points = 128 points, all 128 clusters.
// AoS LDS layout [x,y,z,0] => each WMMA operand fetch is one ds_load_b64
// at (base + 8*half) with no EXEC divergence.
__global__ __launch_bounds__(256) void assign_accum(
    const float* __restrict__ xyz, const float* __restrict__ scores,
    const float* __restrict__ cent,
    float* __restrict__ denom, float* __restrict__ cnum,
    float* __restrict__ gamma_out, int store_gamma)
{
    __shared__ __attribute__((aligned(16))) float s_ct[KK][4];            // cx,cy,cz,0
    __shared__ float s_c2[KK];
    __shared__ __attribute__((aligned(16))) float s_pt[PTS_PER_BLOCK][4]; // px,py,pz,0
    __shared__ float s_p2[PTS_PER_BLOCK], s_w[PTS_PER_BLOCK];
    __shared__ float s_d[KK], s_gx[KK], s_gy[KK], s_gz[KK];

    const int b = blockIdx.y;
    const int nbase = blockIdx.x * PTS_PER_BLOCK;
    const int tid = threadIdx.x;

    if (tid < KK) {
        float cx = cent[(b * KK + tid) * 4 + 0];
        float cy = cent[(b * KK + tid) * 4 + 1];
        float cz = cent[(b * KK + tid) * 4 + 2];
        s_ct[tid][0] = cx; s_ct[tid][1] = cy; s_ct[tid][2] = cz; s_ct[tid][3] = 0.0f;
        s_c2[tid] = cx * cx + cy * cy + cz * cz;
        s_d[tid] = 0.0f; s_gx[tid] = 0.0f; s_gy[tid] = 0.0f; s_gz[tid] = 0.0f;
    } else {
        int p = tid - KK;
        int n = nbase + p;
        const float* xb = xyz + (size_t)b * 3 * NN;
        float px = xb[n], py = xb[NN + n], pz = xb[2 * NN + n];
        s_pt[p][0] = px; s_pt[p][1] = py; s_pt[p][2] = pz; s_pt[p][3] = 0.0f;
        s_p2[p] = px * px + py * py + pz * pz;
        s_w[p] = scores[(size_t)b * NN + n];
    }
    __syncthreads();

    const int wave = tid >> 5;
    const int lane = tid & 31;
    const int half = lane >> 4;
    const int lm   = lane & 15;
    const int pbase = wave * 16;

    // A operand: 16x4 (points x coords); lanes 0-15 hold K=0,1; 16-31 hold K=2,3
    const v2f A = *(const v2f*)&s_pt[pbase + lm][2 * half];

    // per-lane cached point attrs for C-layout rows (M = v + 8*half)
    float p2v[8], wv[8], pxv[8], pyv[8], pzv[8];
#pragma unroll
    for (int v = 0; v < 8; v++) {
        int p = pbase + v + 8 * half;
        p2v[v] = s_p2[p]; wv[v] = s_w[p];
        pxv[v] = s_pt[p][0]; pyv[v] = s_pt[p][1]; pzv[v] = s_pt[p][2];
    }

    float logit[KT][8];
#pragma unroll
    for (int kt = 0; kt < KT; kt++) {
        int c = kt * 16 + lm;
        const v2f Bm = *(const v2f*)&s_ct[c][2 * half];
        v8f acc = {};
        acc = __builtin_amdgcn_wmma_f32_16x16x4_f32(
            false, A, false, Bm, (short)0, acc, false, false);
        float c2 = s_c2[c];
#pragma unroll
        for (int v = 0; v < 8; v++)
            logit[kt][v] = (2.0f * acc[v] - p2v[v] - c2) * (1.0f / TAU);
    }

    // row softmax over 128 clusters: per-lane max/sum over kt, then
    // butterfly over the 16 lanes of each half (xor 1,2,4,8 stays in-half)
    float m[8], ssum[8];
#pragma unroll
    for (int v = 0; v < 8; v++) {
        m[v] = logit[0][v];
#pragma unroll
        for (int kt = 1; kt < KT; kt++) m[v] = fmaxf(m[v], logit[kt][v]);
    }
#pragma unroll
    for (int x = 1; x < 16; x <<= 1)
#pragma unroll
        for (int v = 0; v < 8; v++) m[v] = fmaxf(m[v], __shfl_xor(m[v], x, 32));
#pragma unroll
    for (int v = 0; v < 8; v++) ssum[v] = 0.0f;
#pragma unroll
    for (int kt = 0; kt < KT; kt++)
#pragma unroll
        for (int v = 0; v < 8; v++) {
            float e = __expf(logit[kt][v] - m[v]);
            logit[kt][v] = e;
            ssum[v] += e;
        }
#pragma unroll
    for (int x = 1; x < 16; x <<= 1)
#pragma unroll
        for (int v = 0; v < 8; v++) ssum[v] += __shfl_xor(ssum[v], x, 32);

    float inv[8];
#pragma unroll
    for (int v = 0; v < 8; v++) inv[v] = wv[v] / ssum[v];

#pragma unroll
    for (int kt = 0; kt < KT; kt++) {
        int c = kt * 16 + lm;
        float gs = 0.0f, gx = 0.0f, gy = 0.0f, gz = 0.0f;
#pragma unroll
        for (int v = 0; v < 8; v++) {
            float g = logit[kt][v] * inv[v];
            gs += g; gx += g * pxv[v]; gy += g * pyv[v]; gz += g * pzv[v];
            if (store_gamma) {
                int n = nbase + pbase + v + 8 * half;
                gamma_out[((size_t)b * NN + n) * KK + c] = g;
            }
        }
        gs += __shfl_xor(gs, 16, 32);
        gx += __shfl_xor(gx, 16, 32);
        gy += __shfl_xor(gy, 16, 32);
        gz += __shfl_xor(gz, 16, 32);
        if (half == 0) {
            atomicAdd(&s_d[c], gs);
            atomicAdd(&s_gx[c], gx);
            atomicAdd(&s_gy[c], gy);
            atomicAdd(&s_gz[c], gz);
        }
    }
    __syncthreads();

    if (tid < KK) {
        atomicAdd(&denom[b * KK + tid], s_d[tid]);
        atomicAdd(&cnum[(b * KK + tid) * 3 + 0], s_gx[tid]);
        atomicAdd(&cnum[(b * KK + tid) * 3 + 1], s_gy[tid]);
        atomicAdd(&cnum[(b * KK + tid) * 3 + 2], s_gz[tid]);
    }
}

__global__ void update_centers(const float* __restrict__ denom,
                               const float* __restrict__ cnum,
                               float* __restrict__ cent) {
    int t = blockIdx.x * blockDim.x + threadIdx.x;
    if (t >= BB * KK) return;
    float d = denom[t] + EPSV;
    cent[t * 4 + 0] = cnum[t * 3 + 0] / d;
    cent[t * 4 + 1] = cnum[t * 3 + 1] / d;
    cent[t * 4 + 2] = cnum[t * 3 + 2] / d;
}

__global__ void write_pi_xyz(const float* __restrict__ denom,
                             const float* __restrict__ ssum,
                             const float* __restrict__ cent,
                             float* __restrict__ out) {
    int t = blockIdx.x * blockDim.x + threadIdx.x;
    if (t >= BB * KK) return;
    int b = t / KK;
    float* pi = out + (size_t)BB * NN * KK;
    float* nx = pi + BB * KK;
    pi[t] = denom[t] / ssum[b];
    nx[t * 3 + 0] = cent[t * 4 + 0];
    nx[t * 3 + 1] = cent[t * 4 + 1];
    nx[t * 3 + 2] = cent[t * 4 + 2];
}

// node_feats partial GEMM: out(K x D) = gamma^T (K x N) . feats^T (N x D).
// gamma is read ONCE (4 live accumulators cover all of D=64); feats staged
// through LDS via gfx1250 async global->LDS copies, B operands = ds_load_b64.
__global__ __launch_bounds__(256) void feat_gemm(
    const float* __restrict__ gamma, const float* __restrict__ feats,
    float* __restrict__ facc) {
    __shared__ __attribute__((aligned(16))) float s_f[DD * SUBNP]; // ~33 KB

    const int b = blockIdx.y;
    const int n0base = blockIdx.x * CHUNK;
    const int tid  = threadIdx.x;
    const int wave = tid >> 5;
    const int lane = tid & 31;
    const int half = lane >> 4;
    const int lm   = lane & 15;
    const int kbase = wave * 16;   // 8 waves cover K = 128

    const float* gb = gamma + (size_t)b * NN * KK;
    const float* fb = feats + (size_t)b * DD * NN;

    v8f acc0 = {}, acc1 = {}, acc2 = {}, acc3 = {};

    for (int n0 = n0base; n0 < n0base + CHUNK; n0 += SUBN) {
        __syncthreads();   // previous tile fully consumed
        // stage feats[d][n0 .. n0+SUBN) -> LDS, coalesced 16B per lane
#pragma unroll
        for (int j = 0; j < (DD * SUBN / 4) / 256; j++) {  // 8 iters
            int g  = tid + j * 256;
            int d  = g >> 5;            // SUBN/4 = 32 float4 per row
            int nc = (g & 31) * 4;
            const float* src = fb + (size_t)d * NN + n0 + nc;
            float* dst = &s_f[d * SUBNP + nc];
#ifdef HAVE_ASYNC_LDS
            __builtin_amdgcn_global_load_async_to_lds_b128(
                (GLOBAL_AS v4i*)src, (LDS_AS v4i*)dst, 0, 0);
#else
            *(float4*)dst = *(const float4*)src;
#endif
        }
#ifdef HAVE_ASYNC_LDS
#if __has_builtin(__builtin_amdgcn_s_wait_asynccnt)
        __builtin_amdgcn_s_wait_asynccnt(0);
#else
        asm volatile("s_wait_asynccnt 0" ::: "memory");
#endif
#endif
        __syncthreads();

        for (int nl = 0; nl < SUBN; nl += 4) {
            const int n = n0 + nl;
            v2f A;   // 16 clusters x 4 n's
            A.x = gb[(size_t)(n + 2 * half + 0) * KK + kbase + lm];
            A.y = gb[(size_t)(n + 2 * half + 1) * KK + kbase + lm];
            const v2f B0 = *(const v2f*)&s_f[(0 * 16 + lm) * SUBNP + nl + 2 * half];
            const v2f B1 = *(const v2f*)&s_f[(1 * 16 + lm) * SUBNP + nl + 2 * half];
            const v2f B2 = *(const v2f*)&s_f[(2 * 16 + lm) * SUBNP + nl + 2 * half];
            const v2f B3 = *(const v2f*)&s_f[(3 * 16 + lm) * SUBNP + nl + 2 * half];
            acc0 = __builtin_amdgcn_wmma_f32_16x16x4_f32(false, A, false, B0, (short)0, acc0, false, false);
            acc1 = __builtin_amdgcn_wmma_f32_16x16x4_f32(false, A, false, B1, (short)0, acc1, false, false);
            acc2 = __builtin_amdgcn_wmma_f32_16x16x4_f32(false, A, false, B2, (short)0, acc2, false, false);
            acc3 = __builtin_amdgcn_wmma_f32_16x16x4_f32(false, A, false, B3, (short)0, acc3, false, false);
        }
    }

#pragma unroll
    for (int v = 0; v < 8; v++) {
        int k = kbase + v + 8 * half;
        float* dst = &facc[((size_t)b * KK + k) * DD + lm];
        atomicAdd(dst + 0,  acc0[v]);
        atomicAdd(dst + 16, acc1[v]);
        atomicAdd(dst + 32, acc2[v]);
        atomicAdd(dst + 48, acc3[v]);
    }
}

__global__ void feat_norm(const float* __restrict__ facc,
                          const float* __restrict__ denom,
                          float* __restrict__ out) {
    int t = blockIdx.x * blockDim.x + threadIdx.x;
    if (t >= BB * KK * DD) return;
    int bk = t / DD;
    float* nf = out + (size_t)BB * NN * KK + BB * KK + (size_t)BB * KK * 3;
    nf[t] = facc[t] / (denom[bk] + EPSV);
}

extern "C" void kernel_launch(void* const* d_in, const int* in_sizes, int n_in,
                              void* d_out, int out_size, void* d_ws, size_t ws_size,
                              hipStream_t stream) {
    const float* xyz    = (const float*)d_in[0];
    const float* feats  = (const float*)d_in[1];
    const float* scores = (const float*)d_in[2];
    float* out = (float*)d_out;
    float* ws  = (float*)d_ws;

    float* denom = ws + WS_DENOM;
    float* cnum  = ws + WS_CNUM;
    float* cent  = ws + WS_CENT;
    float* ssum  = ws + WS_SSUM;
    float* facc  = ws + WS_FACC;

    score_sum<<<BB, 256, 0, stream>>>(scores, ssum);
    init_centers<<<(BB * KK + 255) / 256, 256, 0, stream>>>(xyz, cent);

    for (int it = 0; it < ITERS; it++) {
        // denom (B*K) and cnum (B*K*3) are contiguous: zero both in one shot
        zero_f32<<<(BB * KK * 4 + 255) / 256, 256, 0, stream>>>(denom, BB * KK * 4);
        assign_accum<<<dim3(NN / PTS_PER_BLOCK, BB), 256, 0, stream>>>(
            xyz, scores, cent, denom, cnum, out, it == ITERS - 1 ? 1 : 0);
        update_centers<<<(BB * KK + 255) / 256, 256, 0, stream>>>(denom, cnum, cent);
    }

    write_pi_xyz<<<(BB * KK + 255) / 256, 256, 0, stream>>>(denom, ssum, cent, out);

    zero_f32<<<(BB * KK * DD + 255) / 256, 256, 0, stream>>>(facc, BB * KK * DD);
    feat_gemm<<<dim3(NCHUNK, BB), 256, 0, stream>>>(out /* gamma */, feats, facc);
    feat_norm<<<(BB * KK * DD + 255) / 256, 256, 0, stream>>>(facc, denom, out);
}